// WaveletDecomposition_80564996538672
// MI455X (gfx1250) — compile-verified
//
#include <hip/hip_runtime.h>

// ---------------------------------------------------------------------------
// 2-level db4 DWT (symmetric), fp32, via V_WMMA_F32_16X16X4_F32.
//   y[o] = sum_j w[j] * x_sym[2o - 6 + j],  w[j] = DEC[7-j]
//   x_sym: j<0 -> x[-1-j],  j>=L -> x[2L-1-j]
// Tile: 16 outputs (M) x 16 channels (N), K = input positions (40 = 10x4).
//   A[m,t] = w[t - 2m]  (banded, lane-constant)   B[t,n] = x_sym[2*o0-6+t, ch0+n]
// Each wave processes a strip of TPW adjacent tiles: A-fragments built once,
// B slices 8,9 of tile i rotate into slices 0,1 of tile i+1 (8-position
// window overlap), so steady-state tiles do 16 loads + 20 WMMAs + 16 stores.
// Cache policy: NT loads for streamed input, NT stores for final outputs,
// RT stores for cA1 so it stays resident in the 192MB L2 for level 2.
// ---------------------------------------------------------------------------

typedef __attribute__((ext_vector_type(2))) float v2f;
typedef __attribute__((ext_vector_type(8))) float v8f;

__device__ __constant__ float DEC_LO_C[8] = {
    -0.010597401784997278f,  0.032883011666982945f,  0.030841381835986965f,
    -0.18703481171888114f,  -0.02798376941698385f,   0.6308807679295904f,
     0.7148465705525415f,    0.23037781330885523f};
__device__ __constant__ float DEC_HI_C[8] = {
    -0.23037781330885523f,   0.7148465705525415f,   -0.6308807679295904f,
    -0.02798376941698385f,   0.18703481171888114f,   0.030841381835986965f,
    -0.032883011666982945f, -0.010597401784997278f};

#define CHN   256
#define WAVES 8
#define TPW   4   // tiles (16 outputs each) per wave

// Branchless symmetric reflect for p in [-L, 2L-1): select, not exec-mask.
__device__ __forceinline__ int reflect_idx(int p, int L) {
  int r = p ^ (p >> 31);              // p<0 ? -1-p : p
  return (r < L) ? r : (2 * L - 1 - r);
}

template <bool NTA>
__global__ __launch_bounds__(256) void dwt_db4_wmma(
    const float* __restrict__ in, float* __restrict__ outA,
    float* __restrict__ outD, int L, int Lout)
{
  const int lane = threadIdx.x & 31;
  const int wave = threadIdx.x >> 5;
  const int m    = lane & 15;   // M (outputs) for A-frag / N (channel) for B,D
  const int g    = lane >> 4;   // lane-group: K-pair select / D row offset
  const int o_first = ((blockIdx.x * WAVES + wave) * TPW) * 16;
  if (o_first >= Lout) return;  // wave-uniform: EXEC stays all-1s for WMMA
  const int ch    = (blockIdx.y << 4) + m;
  const int batch = blockIdx.z;

  // Lane-constant banded filter fragments, 10 K-slices of 4 (built once,
  // reused for all TPW tiles of this wave).
  v2f alo[10], ahi[10];
#pragma unroll
  for (int s = 0; s < 10; ++s) {
    const int t0 = 4 * s + 2 * g;
    const int i0 = t0 - 2 * m;
    const int i1 = i0 + 1;
    alo[s].x = (i0 >= 0 && i0 < 8) ? DEC_LO_C[7 - i0] : 0.0f;
    alo[s].y = (i1 >= 0 && i1 < 8) ? DEC_LO_C[7 - i1] : 0.0f;
    ahi[s].x = (i0 >= 0 && i0 < 8) ? DEC_HI_C[7 - i0] : 0.0f;
    ahi[s].y = (i1 >= 0 && i1 < 8) ? DEC_HI_C[7 - i1] : 0.0f;
  }

  const float* inb = in + (size_t)batch * L * CHN;
  v2f bf[10];

#pragma unroll
  for (int i = 0; i < TPW; ++i) {
    const int o0 = o_first + 16 * i;
    if (o0 >= Lout) break;                          // uniform per wave
    const int pbase = 2 * o0 - 6 + 2 * g;           // first K position, this lane
    const int s_lo  = (i == 0) ? 0 : 2;             // slices 0,1 rotated in

    // ---- Gather B fragments (NT: input is streamed) -----------------------
    if (2 * o0 - 6 >= 0 && 2 * o0 + 33 < L) {       // interior: imm offsets
      const float* base = inb + (size_t)pbase * CHN + ch;
#pragma unroll
      for (int s = s_lo; s < 10; ++s) {
        bf[s].x = __builtin_nontemporal_load(&base[(size_t)(4 * s)     * CHN]);
        bf[s].y = __builtin_nontemporal_load(&base[(size_t)(4 * s + 1) * CHN]);
      }
    } else {                                        // boundary: reflect (rare)
#pragma unroll
      for (int s = s_lo; s < 10; ++s) {
        const int r0 = reflect_idx(pbase + 4 * s,     L);
        const int r1 = reflect_idx(pbase + 4 * s + 1, L);
        bf[s].x = __builtin_nontemporal_load(&inb[(size_t)r0 * CHN + ch]);
        bf[s].y = __builtin_nontemporal_load(&inb[(size_t)r1 * CHN + ch]);
      }
    }

    // ---- 20 WMMAs ---------------------------------------------------------
    v8f accA = {};  // low-pass  -> cA
    v8f accD = {};  // high-pass -> cD
#pragma unroll
    for (int s = 0; s < 10; ++s) {
      accA = __builtin_amdgcn_wmma_f32_16x16x4_f32(false, alo[s], false, bf[s],
                                                   (short)0, accA, false, false);
      accD = __builtin_amdgcn_wmma_f32_16x16x4_f32(false, ahi[s], false, bf[s],
                                                   (short)0, accD, false, false);
    }

    // ---- Stores -----------------------------------------------------------
    float* oa = outA + ((size_t)batch * Lout + o0 + 8 * g) * CHN + ch;
    float* od = outD + ((size_t)batch * Lout + o0 + 8 * g) * CHN + ch;
    if (o0 + 16 <= Lout) {
#pragma unroll
      for (int v = 0; v < 8; ++v) {
        if (NTA) __builtin_nontemporal_store(accA[v], &oa[(size_t)v * CHN]);
        else     oa[(size_t)v * CHN] = accA[v];     // cA1: keep L2-resident
        __builtin_nontemporal_store(accD[v], &od[(size_t)v * CHN]);
      }
    } else {
#pragma unroll
      for (int v = 0; v < 8; ++v) {
        if (o0 + 8 * g + v < Lout) {
          if (NTA) __builtin_nontemporal_store(accA[v], &oa[(size_t)v * CHN]);
          else     oa[(size_t)v * CHN] = accA[v];
          __builtin_nontemporal_store(accD[v], &od[(size_t)v * CHN]);
        }
      }
    }

    // ---- Rotate shared window: tile i slices 8,9 == tile i+1 slices 0,1 ---
    bf[0] = bf[8];
    bf[1] = bf[9];
  }
}

extern "C" void kernel_launch(void* const* d_in, const int* in_sizes, int n_in,
                              void* d_out, int out_size, void* d_ws, size_t ws_size,
                              hipStream_t stream) {
  (void)in_sizes; (void)n_in; (void)out_size; (void)ws_size;
  const float* x  = (const float*)d_in[0];
  float* out = (float*)d_out;
  float* cA1 = (float*)d_ws;     // 8*8195*256 floats = 67.1 MB scratch (fits L2)

  const int B = 8, S = 16384, C = 256;
  const int L1 = (S + 7) / 2;    // 8195
  const int L2 = (L1 + 7) / 2;   // 4101

  float* cD1 = out;
  float* cD2 = out + (size_t)B * L1 * C;
  float* cA2 = cD2 + (size_t)B * L2 * C;

  const int span = WAVES * TPW * 16;   // outputs covered per block
  dim3 blk(256);
  dim3 g1((unsigned)((L1 + span - 1) / span), (unsigned)(C / 16), (unsigned)B);
  dwt_db4_wmma<false><<<g1, blk, 0, stream>>>(x, cA1, cD1, S, L1);

  dim3 g2((unsigned)((L2 + span - 1) / span), (unsigned)(C / 16), (unsigned)B);
  dwt_db4_wmma<true><<<g2, blk, 0, stream>>>(cA1, cA2, cD2, L1, L2);
}